// SoftNearestNeighbours_63324997812618
// MI455X (gfx1250) — compile-verified
//
#include <hip/hip_runtime.h>
#include <hip/hip_bf16.h>

typedef __attribute__((ext_vector_type(2))) float v2f;
typedef __attribute__((ext_vector_type(4))) float v4f;
typedef __attribute__((ext_vector_type(8))) float v8f;

#define N_PTS 8192
#define DIMS  64
#define TILES (N_PTS / 16)         // 512 column tiles
#define WAVES 4                    // waves per block (128 threads, wave32)
#define TPW   (TILES / WAVES)      // 128 contiguous tiles per wave
#define TILE_BYTES 4096            // 16 rows x 64 f32, contiguous in x

// One async b128 = 32 lanes x 16B = 512B; 8 cover a 4KB tile.
#define ASYNC_LDS_B128(lds, gaddr, OFF)                                   \
    asm volatile("global_load_async_to_lds_b128 %0, %1, off offset:" OFF  \
                 :: "v"(lds), "v"(gaddr) : "memory")

__device__ __forceinline__ void prefetch_tile(uint32_t lds, unsigned long long g) {
    ASYNC_LDS_B128(lds, g, "0");
    ASYNC_LDS_B128(lds, g, "512");
    ASYNC_LDS_B128(lds, g, "1024");
    ASYNC_LDS_B128(lds, g, "1536");
    ASYNC_LDS_B128(lds, g, "2048");
    ASYNC_LDS_B128(lds, g, "2560");
    ASYNC_LDS_B128(lds, g, "3072");
    ASYNC_LDS_B128(lds, g, "3584");
}

// ---------------------------------------------------------------------------
// Kernel 1: per-row squared norms into workspace; zero the scalar output.
// ---------------------------------------------------------------------------
__global__ void snnl_setup(const float* __restrict__ x,
                           float* __restrict__ sq,
                           float* __restrict__ out) {
    int i = blockIdx.x * blockDim.x + threadIdx.x;
    if (i == 0) out[0] = 0.0f;
    if (i < N_PTS) {
        const v4f* row = (const v4f*)(x + (size_t)i * DIMS);
        float s = 0.0f;
#pragma unroll
        for (int k = 0; k < DIMS / 4; ++k) {
            v4f v = row[k];
            s += v.x * v.x + v.y * v.y + v.z * v.z + v.w * v.w;
        }
        sq[i] = s;
    }
}

// ---------------------------------------------------------------------------
// Kernel 2: one block (128 thr = 4 waves) per 16-row tile. Each wave sweeps
// 128 contiguous column tiles. B tiles are DMA'd into a per-wave LDS double
// buffer with GLOBAL_LOAD_ASYNC_TO_LDS_B128 (ASYNCcnt), prefetched one tile
// ahead; WMMAs read B from LDS (ds_load_b64). Exact-f32 Gram via chained
// V_WMMA_F32_16X16X4_F32; fused sqrt/exp/mask epilogue; shuffle + LDS reduce.
// ---------------------------------------------------------------------------
__global__ __launch_bounds__(128) void snnl_main(const float* __restrict__ x,
                                                 const int*   __restrict__ y,
                                                 const float* __restrict__ wptr,
                                                 const float* __restrict__ sq,
                                                 float* __restrict__ out) {
    __shared__ float bstage[WAVES][2][TILE_BYTES / 4];   // 32 KB staging
    __shared__ float top_s[16];
    __shared__ float bot_s[16];

    const int lane    = threadIdx.x & 31;
    const int wave    = threadIdx.x >> 5;
    const int rowBase = blockIdx.x * 16;
    const int lr      = lane & 15;   // row/col index within a 16-wide tile
    const int hh      = lane >> 4;   // which 16-lane half (0 or 1)

    if (threadIdx.x < 16) { top_s[threadIdx.x] = 0.0f; bot_s[threadIdx.x] = 0.0f; }
    __syncthreads();

    const float negw = -wptr[0];

    // A tile resident: lane holds x[rowBase+lr][4*kk + 2*hh .. +1]  (ISA A 16x4 layout)
    v2f a[16];
#pragma unroll
    for (int kk = 0; kk < 16; ++kk)
        a[kk] = *(const v2f*)(x + (size_t)(rowBase + lr) * DIMS + 4 * kk + 2 * hh);

    // Per-lane metadata for the C/D layout rows: m = rowBase + r + 8*hh
    float sqm[8];
    int   ym[8];
#pragma unroll
    for (int r = 0; r < 8; ++r) {
        int m  = rowBase + r + 8 * hh;
        sqm[r] = sq[m];
        ym[r]  = y[m];
    }

    float topAcc[8], botAcc[8];
#pragma unroll
    for (int r = 0; r < 8; ++r) { topAcc[r] = 0.0f; botAcc[r] = 0.0f; }

    // Per-lane LDS targets (low 32 bits of generic shared address = LDS offset)
    const uint32_t ldsA = (uint32_t)(uintptr_t)&bstage[wave][0][0] + (uint32_t)lane * 16u;
    const uint32_t ldsB = ldsA + TILE_BYTES;

    const int startTile = wave * TPW;
    // Per-lane global source for tile t: x + t*4096 + lane*16 bytes
    const unsigned long long g0 =
        (unsigned long long)(uintptr_t)x +
        (unsigned long long)startTile * TILE_BYTES + (unsigned long long)lane * 16u;

    // Prologue: DMA tile 0 of this wave's range into buffer 0.
    prefetch_tile(ldsA, g0);

    for (int j = 0; j < TPW; ++j) {
        // Prefetch next tile into the other buffer (last iter: dummy reload of
        // tile 0 so the asynccnt<=8 wait below always has 8 newer ops behind it).
        const int tnext = (j + 1 < TPW) ? (j + 1) : 0;
        prefetch_tile((j & 1) ? ldsA : ldsB,
                      g0 + (unsigned long long)tnext * TILE_BYTES);

        // Async loads complete in order: <=8 outstanding => tile j has landed.
        asm volatile("s_wait_asynccnt 0x8" ::: "memory");

        const float* bt = &bstage[wave][j & 1][0];
        const int colBase = (startTile + j) * 16;

        const float sqn = sq[colBase + lr];
        const int   yn  = y[colBase + lr];
        const int   n   = colBase + lr;

        // B from LDS in the ISA 4x16 B layout; exact-f32 Gram via 16 WMMAs.
        v8f c = {};
#pragma unroll
        for (int kk = 0; kk < 16; ++kk) {
            v2f b = *(const v2f*)&bt[lr * DIMS + 4 * kk + 2 * hh];
            c = __builtin_amdgcn_wmma_f32_16x16x4_f32(
                    false, a[kk], false, b, (short)0, c, false, false);
        }

        // Fused epilogue on the 8 elements this lane owns
#pragma unroll
        for (int r = 0; r < 8; ++r) {
            const int m    = rowBase + r + 8 * hh;
            float     d2   = sqm[r] + sqn - 2.0f * c[r];
            d2             = fmaxf(d2, 0.0f);
            float     e    = __expf(negw * __builtin_sqrtf(d2));
            e              = (m == n) ? 0.0f : e;   // exclude diagonal (top & bot)
            botAcc[r] += e;
            topAcc[r] += (ym[r] == yn) ? e : 0.0f;
        }
    }

    // Row sums: xor-shuffle reduce across the 16 lanes of each half,
    // then one LDS float-atomic per row per wave.
#pragma unroll
    for (int r = 0; r < 8; ++r) {
        float t = topAcc[r];
        float bsum = botAcc[r];
#pragma unroll
        for (int mask = 1; mask <= 8; mask <<= 1) {
            t    += __shfl_xor(t, mask, 32);
            bsum += __shfl_xor(bsum, mask, 32);
        }
        if (lr == 0) {
            atomicAdd(&top_s[r + 8 * hh], t);
            atomicAdd(&bot_s[r + 8 * hh], bsum);
        }
    }
    __syncthreads();

    // loss contribution: -(1/N) * log(top/(bot+eps)) per row, atomically summed
    if (threadIdx.x < 16) {
        float frac = logf(top_s[threadIdx.x] / (bot_s[threadIdx.x] + 1e-8f));
        atomicAdd(out, -frac * (1.0f / (float)N_PTS));
    }
}

// ---------------------------------------------------------------------------
extern "C" void kernel_launch(void* const* d_in, const int* in_sizes, int n_in,
                              void* d_out, int out_size, void* d_ws, size_t ws_size,
                              hipStream_t stream) {
    const float* x = (const float*)d_in[0];
    const int*   y = (const int*)d_in[1];
    const float* w = (const float*)d_in[2];
    float*   out   = (float*)d_out;
    float*   sq    = (float*)d_ws;   // 8192 floats = 32 KB scratch

    snnl_setup<<<(N_PTS + 255) / 256, 256, 0, stream>>>(x, sq, out);
    snnl_main<<<N_PTS / 16, 128, 0, stream>>>(x, y, w, sq, out);
}